// BaseRGCN_57088705298757
// MI455X (gfx1250) — compile-verified
//
#include <hip/hip_runtime.h>

#define N_NODES 100000
#define N_EDGES 1600000
#define H_DIM   64
#define OUT_DIM 64
#define N_RELS  16
#define N_BASES 4

typedef __attribute__((ext_vector_type(2))) float v2f;
typedef __attribute__((ext_vector_type(8))) float v8f;

// ---------------------------------------------------------------- zero fill
__global__ void rgcn_zero_f32(float* __restrict__ p, long n) {
    long i = (long)blockIdx.x * blockDim.x + threadIdx.x;
    if (i < n) p[i] = 0.0f;
}

// ---------------------------------------------------------------- edge gather + scatter-add
// msg[e, f] = sum_b wc[r(e), b] * V[b, src(e), f];  atomicAdd into agg[dst(e), f]
// V laid out [N_BASES, N_NODES, 64]. 32 lanes per edge, float2 per lane.
__global__ void __launch_bounds__(256)
rgcn_edge_agg(const int*   __restrict__ src,
              const int*   __restrict__ dst,
              const int*   __restrict__ et,
              const float* __restrict__ V,    // [N_BASES, N_NODES, 64]
              const float* __restrict__ wc,   // [N_RELS, N_BASES]
              float*       __restrict__ agg)  // [N_NODES, 64]
{
    const int  t = threadIdx.x;
    const long e = (long)blockIdx.x * 8 + (t >> 5);
    if (e >= N_EDGES) return;
    const int lane = t & 31;
    const int f    = lane << 1;              // feature pair

    const int s = src[e];
    const int d = dst[e];
    const int r = et[e];
    const float w0 = wc[r * 4 + 0];
    const float w1 = wc[r * 4 + 1];
    const float w2 = wc[r * 4 + 2];
    const float w3 = wc[r * 4 + 3];

    const size_t bs   = (size_t)N_NODES * H_DIM;
    const size_t base = (size_t)s * H_DIM + f;
    v2f m0 = *(const v2f*)(V + base);
    v2f m1 = *(const v2f*)(V + bs + base);
    v2f m2 = *(const v2f*)(V + 2 * bs + base);
    v2f m3 = *(const v2f*)(V + 3 * bs + base);
    v2f m  = w0 * m0 + w1 * m1 + w2 * m2 + w3 * m3;

    float* out = agg + (size_t)d * H_DIM + f;
    atomicAdd(out,     m.x);
    atomicAdd(out + 1, m.y);
}

// ---------------------------------------------------------------- h = relu(agg1 + loop1[feats] + bias1)
__global__ void rgcn_hidden_relu(const float* __restrict__ agg1,
                                 const int*   __restrict__ feats,
                                 const float* __restrict__ loop1,
                                 const float* __restrict__ bias1,
                                 float*       __restrict__ h)
{
    long i = (long)blockIdx.x * blockDim.x + threadIdx.x;
    if (i >= (long)N_NODES * H_DIM) return;
    const int n = (int)(i >> 6);
    const int f = (int)(i & 63);
    float v = agg1[i] + loop1[(size_t)feats[n] * H_DIM + f] + bias1[f];
    h[i] = v > 0.0f ? v : 0.0f;
}

// ---------------------------------------------------------------- fused GEMM via V_WMMA_F32_16X16X4_F32
// Y = h[100000 x 64] @ W[64 x 320], W columns: [V2[0] | V2[1] | V2[2] | V2[3] | loop2].
// Tiles 0..15 -> HB[b, :, :]   (basis projections, layout [4, N, 64])
// Tiles 16..19 -> agg2 (initialized with h @ loop2; edges atomically add on top later)
// One wave per 16x16 output tile. 16 k-steps of K=4, split across 2 accumulators.
__global__ void __launch_bounds__(128)
rgcn_wmma_gemm(const float* __restrict__ h,      // [N, 64]
               const float* __restrict__ V2,     // [4, 64, 64]
               const float* __restrict__ loop2,  // [64, 64]
               float*       __restrict__ HB,     // [4, N, 64]
               float*       __restrict__ agg2)   // [N, 64]
{
    const int wave   = threadIdx.x >> 5;
    const int lane   = threadIdx.x & 31;
    const int lane16 = lane & 15;
    const int half   = lane >> 4;            // 0: lanes 0-15 (K=0,1) / 1: lanes 16-31 (K=2,3)
    const int ct     = blockIdx.y * 4 + wave; // column tile 0..19 (wave-uniform)
    const int tn     = ct << 4;               // first output column of tile
    const int row0   = blockIdx.x << 4;       // 100000 / 16 = 6250 exact row tiles
    const size_t N64 = (size_t)N_NODES * H_DIM;

    const float* bp;   // B operand base: element [k=0][col = tn%64 + lane16]
    float*       wdst; // output surface
    if (tn < 256) {
        bp   = V2 + (size_t)(tn >> 6) * (H_DIM * OUT_DIM) + (tn & 63) + lane16;
        wdst = HB + (size_t)(tn >> 6) * N64;
    } else {
        bp   = loop2 + (tn & 63) + lane16;
        wdst = agg2;
    }
    // A operand: row = row0 + lane16, K pair starting at 2*half (per ISA 16x4 f32 layout)
    const float* ap = h + (size_t)(row0 + lane16) * H_DIM + (half << 1);
    const int kofs  = half << 1;

    v8f acc0 = {};
    v8f acc1 = {};
#pragma unroll
    for (int k = 0; k < H_DIM; k += 8) {
        v2f a0 = *(const v2f*)(ap + k);
        v2f b0;
        b0.x = bp[(k + kofs)     * OUT_DIM];
        b0.y = bp[(k + kofs + 1) * OUT_DIM];
        acc0 = __builtin_amdgcn_wmma_f32_16x16x4_f32(false, a0, false, b0,
                                                     (short)0, acc0, false, false);
        v2f a1 = *(const v2f*)(ap + k + 4);
        v2f b1;
        b1.x = bp[(k + 4 + kofs)     * OUT_DIM];
        b1.y = bp[(k + 4 + kofs + 1) * OUT_DIM];
        acc1 = __builtin_amdgcn_wmma_f32_16x16x4_f32(false, a1, false, b1,
                                                     (short)0, acc1, false, false);
    }
    v8f acc = acc0 + acc1;

    // C/D layout: VGPR v, lanes 0-15 -> M=v, lanes 16-31 -> M=v+8; N = lane16
#pragma unroll
    for (int v = 0; v < 8; ++v) {
        const int m = v + (half << 3);
        wdst[(size_t)(row0 + m) * OUT_DIM + (tn & 63) + lane16] = acc[v];
    }
}

// ---------------------------------------------------------------- out = relu(agg2 + bias2)
__global__ void rgcn_out_relu(const float* __restrict__ agg2,
                              const float* __restrict__ bias2,
                              float*       __restrict__ out)
{
    long i = (long)blockIdx.x * blockDim.x + threadIdx.x;
    if (i >= (long)N_NODES * OUT_DIM) return;
    float v = agg2[i] + bias2[i & 63];
    out[i] = v > 0.0f ? v : 0.0f;
}

// ---------------------------------------------------------------- launch
extern "C" void kernel_launch(void* const* d_in, const int* in_sizes, int n_in,
                              void* d_out, int out_size, void* d_ws, size_t ws_size,
                              hipStream_t stream) {
    const int*   src   = (const int*)  d_in[0];
    const int*   dst   = (const int*)  d_in[1];
    const int*   et    = (const int*)  d_in[2];
    const int*   feats = (const int*)  d_in[3];
    const float* V1    = (const float*)d_in[4];
    const float* wc1   = (const float*)d_in[5];
    const float* loop1 = (const float*)d_in[6];
    const float* bias1 = (const float*)d_in[7];
    const float* V2    = (const float*)d_in[8];
    const float* wc2   = (const float*)d_in[9];
    const float* loop2 = (const float*)d_in[10];
    const float* bias2 = (const float*)d_in[11];
    float*       out   = (float*)d_out;
    float*       ws    = (float*)d_ws;

    const size_t N64 = (size_t)N_NODES * H_DIM;
    float* h    = ws;             // [N, 64]
    float* HB   = ws + N64;       // [4, N, 64]; first slice doubles as agg1 (dead before GEMM)
    float* agg1 = HB;             // [N, 64]
    float* agg2 = ws + 5 * N64;   // [N, 64]; fully written by GEMM (h @ loop2)

    // layer 1: zero agg1, edge gather/scatter, self-loop + bias + relu -> h
    rgcn_zero_f32<<<(int)((N64 + 255) / 256), 256, 0, stream>>>(agg1, (long)N64);
    rgcn_edge_agg<<<N_EDGES / 8, 256, 0, stream>>>(src, dst, et, V1, wc1, agg1);
    rgcn_hidden_relu<<<(int)((N64 + 255) / 256), 256, 0, stream>>>(agg1, feats, loop1, bias1, h);

    // layer 2: fused WMMA GEMM -> HB (basis projections) + agg2 (= h @ loop2),
    // then edge gather/scatter on HB adds messages into agg2, then bias + relu.
    rgcn_wmma_gemm<<<dim3(N_NODES / 16, 5), 128, 0, stream>>>(h, V2, loop2, HB, agg2);
    rgcn_edge_agg<<<N_EDGES / 8, 256, 0, stream>>>(src, dst, et, HB, wc2, agg2);
    rgcn_out_relu<<<(int)((N64 + 255) / 256), 256, 0, stream>>>(agg2, bias2, out);
}